// SimpleRGCNOld_15547781611629
// MI455X (gfx1250) — compile-verified
//
#include <hip/hip_runtime.h>
#include <hip/hip_bf16.h>

// Problem constants (match reference)
#define BB   8
#define NN   4096
#define EE   131072
#define EMB  128
#define RR   16
#define KTOT (RR * EMB)   // 2048

typedef __attribute__((ext_vector_type(2))) float v2f;
typedef __attribute__((ext_vector_type(8))) float v8f;

// ---------------------------------------------------------------------------
// Phase 0: zero the per-batch workspace (agg slice + deg slice), float4 wide.
// ---------------------------------------------------------------------------
__global__ __launch_bounds__(256) void rgcn_zero(float4* __restrict__ p, int n4) {
    int i = blockIdx.x * 256 + threadIdx.x;
    const int stride = gridDim.x * 256;
    const float4 z = {0.f, 0.f, 0.f, 0.f};
    for (; i < n4; i += stride) p[i] = z;
}

// ---------------------------------------------------------------------------
// Phase 1: per-batch row degrees. row = rel*N + src  (float counts, exact).
// ---------------------------------------------------------------------------
__global__ __launch_bounds__(256) void rgcn_deg(const int* __restrict__ esrc,
                                                const int* __restrict__ erel,
                                                float* __restrict__ deg) {
    const int e = blockIdx.x * 256 + threadIdx.x;
    if (e < EE) atomicAdd(&deg[erel[e] * NN + esrc[e]], 1.0f);
}

// ---------------------------------------------------------------------------
// Phase 2: scatter UNscaled dst embeddings into agg rows (1/deg folded into
// the GEMM A-staging instead). One wave per edge; lane loads float4 of the
// embedding (coalesced b128) and issues 4 f32 atomic adds (L2-resident).
// ---------------------------------------------------------------------------
__global__ __launch_bounds__(256) void rgcn_scatter(const int* __restrict__ esrc,
                                                    const int* __restrict__ erel,
                                                    const int* __restrict__ edst,
                                                    const float* __restrict__ emb,
                                                    float* __restrict__ agg) {
    const int e    = blockIdx.x * 8 + (threadIdx.x >> 5);
    const int lane = threadIdx.x & 31;
    const int src = esrc[e];
    const int rel = erel[e];
    const int dst = edst[e];
    const int row = rel * NN + src;
    const float4 v = *(const float4*)(emb + (size_t)dst * EMB + lane * 4);
    float* base = agg + (size_t)row * EMB + lane * 4;
    atomicAdd(base + 0, v.x);
    atomicAdd(base + 1, v.y);
    atomicAdd(base + 2, v.z);
    atomicAdd(base + 3, v.w);
}

// ---------------------------------------------------------------------------
// Phase 3: out[n, i] = relu( sum_{k=(r,j)} (agg[r,n,j]/deg[r,n]) * W[r,i,j] )
// GEMM: M = 4096 nodes (16/workgroup), N = 128 cols (16/wave, 8 waves),
// K = 2048 in 64-wide LDS-staged chunks, fp32 WMMA 16x16x4.
//
// LDS tiles are stored as float2 (K,K+1)-pairs so each WMMA operand fetch is
// one ds_load_b64 (2 LDS loads per WMMA instead of 4 scalar loads):
//   A: pair-stride 17  -> stores conflict-free, loads ~2-bank overlap
//   B: pair-stride 137 -> stores conflict-free, loads 14-bank 2-way overlap
// ---------------------------------------------------------------------------
#define ASTRIDE 17
#define BSTRIDE 137

__global__ __launch_bounds__(256) void rgcn_gemm(const float* __restrict__ agg,
                                                 const float* __restrict__ deg,
                                                 const float* __restrict__ w,
                                                 float* __restrict__ out) {
    __shared__ v2f lds_a[32 * ASTRIDE];   //  4352 B
    __shared__ v2f lds_b[32 * BSTRIDE];   // 35072 B

    const int tid  = threadIdx.x;
    const int lane = tid & 31;
    const int wave = tid >> 5;          // 0..7
    const int n0   = blockIdx.x * 16;   // node tile base (M)
    const int i0   = wave * 16;         // output column base (N)
    const int lh   = lane >> 4;         // half-wave: K-pair offset 0 or 1
    const int ln   = lane & 15;

    // A-staging coords: thread t loads A[sm][4 consecutive k] of the chunk
    const int sm = tid >> 4;            // 0..15 (node row within tile)
    const int sp = (tid & 15) * 2;      // first K-pair index (0..30)

    v8f acc = {};                       // 16x16 f32 C/D fragment (8 VGPRs)

    for (int c = 0; c < KTOT / 64; ++c) {        // 32 chunks
        const int r  = c >> 1;                   // relation of this chunk
        const int j0 = (c & 1) * 64;             // j base within EMB

        // ---- stage A (scaled by 1/deg; deg==0 -> row stays zero) ----
        {
            const float d = deg[r * NN + n0 + sm];
            const float s = (d > 0.0f) ? (1.0f / d) : 0.0f;
            const float* ap =
                agg + ((size_t)(r * NN + n0 + sm)) * EMB + j0 + sp * 2;
            const float4 av = *(const float4*)ap;
            // prefetch next chunk's A stream into WGP$ (L2-resident already)
            if (c + 1 < KTOT / 64) {
                const int rn  = (c + 1) >> 1;
                const int j0n = ((c + 1) & 1) * 64;
                __builtin_prefetch(
                    agg + ((size_t)(rn * NN + n0 + sm)) * EMB + j0n + sp * 2, 0, 3);
            }
            v2f p0; p0.x = av.x * s; p0.y = av.y * s;
            v2f p1; p1.x = av.z * s; p1.y = av.w * s;
            lds_a[(sp + 0) * ASTRIDE + sm] = p0;
            lds_a[(sp + 1) * ASTRIDE + sm] = p1;
        }
        // ---- stage B: W[r, i, j0..j0+63] -> lds_b[kpair][i] ----
        #pragma unroll
        for (int it = 0; it < 8; ++it) {
            const int q  = tid + it * 256;       // 0..2047 float4s
            const int bi = q >> 4;               // i: 0..127
            const int bp = (q & 15) * 2;         // first K-pair index
            const float4 bv =
                *(const float4*)(w + ((size_t)(r * EMB + bi)) * EMB + j0 + bp * 2);
            v2f p0; p0.x = bv.x; p0.y = bv.y;
            v2f p1; p1.x = bv.z; p1.y = bv.w;
            lds_b[(bp + 0) * BSTRIDE + bi] = p0;
            lds_b[(bp + 1) * BSTRIDE + bi] = p1;
        }
        __syncthreads();

        // ---- 16 x WMMA f32 16x16x4 over the 64-wide chunk ----
        // A lane layout: lanes 0-15 hold pair (K,K+1), lanes 16-31 (K+2,K+3)
        #pragma unroll
        for (int kk = 0; kk < 16; ++kk) {
            const int pa = kk * 2 + lh;          // K-pair index for this lane
            const v2f a = lds_a[pa * ASTRIDE + ln];
            const v2f b = lds_b[pa * BSTRIDE + i0 + ln];
            acc = __builtin_amdgcn_wmma_f32_16x16x4_f32(
                /*neg_a=*/false, a, /*neg_b=*/false, b,
                /*c_mod=*/(short)0, acc, /*reuse_a=*/false, /*reuse_b=*/false);
        }
        __syncthreads();
    }

    // D layout: VGPR g -> row g (lanes 0-15) / row g+8 (lanes 16-31), col = ln
    #pragma unroll
    for (int g = 0; g < 8; ++g) {
        const int m = g + lh * 8;
        out[((size_t)(n0 + m)) * EMB + i0 + ln] = fmaxf(acc[g], 0.0f);
    }
}

// ---------------------------------------------------------------------------
// Host-side launcher: per-batch pipeline keeps the 32MB agg slice L2-resident.
// ws layout: [agg: R*N*EMB f32 = 32 MB][deg: R*N f32 = 256 KB]
// ---------------------------------------------------------------------------
extern "C" void kernel_launch(void* const* d_in, const int* in_sizes, int n_in,
                              void* d_out, int out_size, void* d_ws, size_t ws_size,
                              hipStream_t stream) {
    (void)in_sizes; (void)n_in; (void)out_size; (void)ws_size;
    // setup_inputs order: batch_nodes, edge_src, edge_rel, edge_dst, embeddings, weights
    const int*   esrc = (const int*)d_in[1];
    const int*   erel = (const int*)d_in[2];
    const int*   edst = (const int*)d_in[3];
    const float* emb  = (const float*)d_in[4];
    const float* wts  = (const float*)d_in[5];
    float*       out  = (float*)d_out;

    float* agg = (float*)d_ws;                         // R*N*EMB floats
    float* deg = agg + (size_t)RR * NN * EMB;          // R*N floats
    const int n4 = (RR * NN * EMB + RR * NN) / 4;      // float4 count to zero

    for (int b = 0; b < BB; ++b) {
        const int*   es = esrc + (size_t)b * EE;
        const int*   er = erel + (size_t)b * EE;
        const int*   ed = edst + (size_t)b * EE;
        const float* eb = emb + (size_t)b * NN * EMB;
        float*       ob = out + (size_t)b * NN * EMB;

        rgcn_zero<<<2048, 256, 0, stream>>>((float4*)agg, n4);
        rgcn_deg<<<EE / 256, 256, 0, stream>>>(es, er, deg);
        rgcn_scatter<<<EE / 8, 256, 0, stream>>>(es, er, ed, eb, agg);
        rgcn_gemm<<<NN / 16, 256, 0, stream>>>(agg, deg, wts, ob);
    }
}